// BoundaryLoss_38027640439294
// MI455X (gfx1250) — compile-verified
//
#include <hip/hip_runtime.h>
#include <hip/hip_bf16.h>

#define B_ 4
#define C_ 19
#define H_ 256
#define W_ 256
#define INF_ 1.0e10f

typedef __attribute__((ext_vector_type(2))) float v2f;
typedef __attribute__((ext_vector_type(8))) float v8f;

// ---------------------------------------------------------------------------
// Pass 1: per-row 1D squared EDT (min-plus with binary sources) along W.
// For slice s=(b,c): tmpA = dist^2 to nearest pixel with target==c (or 1e10),
//                    tmpB = dist^2 to nearest pixel with target!=c (or 1e10).
// Bitwise-identical to reference: (j-j')^2 is exact in fp32 (<= 255^2) and
// when no source exists the reference min is exactly INF (1e10 + 0).
// ---------------------------------------------------------------------------
__global__ __launch_bounds__(256) void bl_pass1(const int* __restrict__ target,
                                                float* __restrict__ tmpA,
                                                float* __restrict__ tmpB) {
  const int bh = blockIdx.x;   // 0 .. B*H-1
  const int b  = bh >> 8;      // H == 256
  const int h  = bh & 255;
  const int j  = threadIdx.x;

  __shared__ int trow[W_];
  int t = target[(size_t)(b * H_ + h) * W_ + j];
  t = t < 0 ? 0 : (t > C_ - 1 ? C_ - 1 : t);   // clip(target, 0, C-1)
  trow[j] = t;
  __syncthreads();

  #pragma unroll 1
  for (int c = 0; c < C_; ++c) {
    float mA = 3.0e38f, mB = 3.0e38f;
    #pragma unroll 4
    for (int jp = 0; jp < W_; ++jp) {
      const int   dj = j - jp;
      const float d2 = (float)(dj * dj);
      const bool  eq = (trow[jp] == c);        // LDS broadcast (uniform addr)
      mA = fminf(mA, eq ? d2 : 3.0e38f);
      mB = fminf(mB, eq ? 3.0e38f : d2);
    }
    const size_t off = ((size_t)(b * C_ + c) * H_ + h) * W_ + j;
    tmpA[off] = (mA > 1.0e9f) ? INF_ : mA;
    tmpB[off] = (mB > 1.0e9f) ? INF_ : mB;
  }
}

// ---------------------------------------------------------------------------
// Pass 2 (fused): column min-plus along H + threshold + sqrt + softmax coef
// + per-pixel class sum + block reduction. One block per (b, w), thread = h.
// dist = sqrt(cleanA) + sqrt(cleanB) equals the reference's masked fp+fn sum
// (at a pos pixel cleanA==0, at a neg pixel cleanB==0; empty-mask slices give
//  INF -> thresholded to 0, matching the has_pos guard).
// ---------------------------------------------------------------------------
__global__ __launch_bounds__(256) void bl_pass2_loss(const float* __restrict__ pred,
                                                     const int*   __restrict__ target,
                                                     const float* __restrict__ tmpA,
                                                     const float* __restrict__ tmpB,
                                                     float*       __restrict__ blockSums) {
  const int bw = blockIdx.x;   // 0 .. B*W-1
  const int b  = bw >> 8;      // W == 256
  const int w  = bw & 255;
  const int h  = threadIdx.x;

  // softmax over C for pixel (b, h, w)
  float lg[C_];
  #pragma unroll
  for (int c = 0; c < C_; ++c)
    lg[c] = pred[((size_t)(b * C_ + c) * H_ + h) * W_ + w];
  float mx = lg[0];
  #pragma unroll
  for (int c = 1; c < C_; ++c) mx = fmaxf(mx, lg[c]);
  float se = 0.0f;
  #pragma unroll
  for (int c = 0; c < C_; ++c) { lg[c] = expf(lg[c] - mx); se += lg[c]; }
  const float inv = 1.0f / se;

  const int traw = target[(size_t)(b * H_ + h) * W_ + w];
  const int tcl  = traw < 0 ? 0 : (traw > C_ - 1 ? C_ - 1 : traw);
  const float valid = (traw != 255) ? 1.0f : 0.0f;

  __shared__ float colA[H_];
  __shared__ float colB[H_];

  float loss = 0.0f;
  #pragma unroll 1
  for (int c = 0; c < C_; ++c) {
    const size_t base = (size_t)(b * C_ + c) * H_ * W_ + w;
    colA[h] = tmpA[base + (size_t)h * W_];
    colB[h] = tmpB[base + (size_t)h * W_];
    if (c + 1 < C_) {  // prefetch next class's column (global_prefetch_b8)
      __builtin_prefetch(tmpA + base + (size_t)H_ * W_ + (size_t)h * W_, 0, 0);
      __builtin_prefetch(tmpB + base + (size_t)H_ * W_ + (size_t)h * W_, 0, 0);
    }
    __syncthreads();

    float m1 = 3.0e38f, m2 = 3.0e38f;
    #pragma unroll 4
    for (int hp = 0; hp < H_; ++hp) {
      const int   dh = h - hp;
      const float d2 = (float)(dh * dh);
      m1 = fminf(m1, colA[hp] + d2);   // LDS broadcast reads
      m2 = fminf(m2, colB[hp] + d2);
    }
    __syncthreads();

    const float gA = (m1 > 0.5f * INF_) ? 0.0f : m1;  // reference threshold
    const float gB = (m2 > 0.5f * INF_) ? 0.0f : m2;
    const float dist = sqrtf(gA) + sqrtf(gB);

    const float p = lg[c] * inv;
    const float t = (tcl == c) ? 1.0f : 0.0f;
    const float coef = valid * (p + t - 2.0f * p * t);  // fps + fns
    loss += coef * dist;
  }

  // deterministic block reduction
  __shared__ float red[256];
  red[h] = loss;
  __syncthreads();
  for (int st = 128; st > 0; st >>= 1) {
    if (h < st) red[h] += red[h + st];
    __syncthreads();
  }
  if (h == 0) blockSums[blockIdx.x] = red[0];
}

// ---------------------------------------------------------------------------
// Final reduction: one wave32. Lane-strided partial sums, then a full-precision
// FP32 WMMA (V_WMMA_F32_16X16X4_F32) with B = ones reduces the 32 lane sums to
// 16 row sums; lanes 0/16 hold the two halves. Deterministic order.
// ---------------------------------------------------------------------------
__global__ __launch_bounds__(32) void bl_reduce(const float* __restrict__ blockSums,
                                                float* __restrict__ out,
                                                int n, float scale) {
  const int lane = threadIdx.x;  // 0..31, EXEC all ones (WMMA requirement)
  float s = 0.0f;
  for (int i = lane; i < n; i += 32) s += blockSums[i];

#if defined(__AMDGCN__) && __has_builtin(__builtin_amdgcn_wmma_f32_16x16x4_f32)
  // A(16x4 f32): vgpr0 = lane sum, vgpr1 = 0  =>  A[m,0]=s_m, A[m,2]=s_{m+16}
  v2f a;  a.x = s;    a.y = 0.0f;
  v2f bo; bo.x = 1.0f; bo.y = 1.0f;      // B(4x16) = all ones
  v8f cz = {};
  v8f d = __builtin_amdgcn_wmma_f32_16x16x4_f32(
      /*neg_a=*/false, a, /*neg_b=*/false, bo,
      /*c_mod=*/(short)0, cz, /*reuse_a=*/false, /*reuse_b=*/false);
  // D[m,n] = s_m + s_{m+16}. Lane 0 holds M=0..7 (col 0), lane 16 holds M=8..15.
  float rs = d[0] + d[1] + d[2] + d[3] + d[4] + d[5] + d[6] + d[7];
  __shared__ float sh2[2];
  if (lane == 0)  sh2[0] = rs;
  if (lane == 16) sh2[1] = rs;
  __syncthreads();
  if (lane == 0) out[0] = (sh2[0] + sh2[1]) * scale;
#else
  __shared__ float sh[32];
  sh[lane] = s;
  __syncthreads();
  if (lane == 0) {
    float tot = 0.0f;
    for (int i = 0; i < 32; ++i) tot += sh[i];
    out[0] = tot * scale;
  }
#endif
}

extern "C" void kernel_launch(void* const* d_in, const int* in_sizes, int n_in,
                              void* d_out, int out_size, void* d_ws, size_t ws_size,
                              hipStream_t stream) {
  (void)in_sizes; (void)n_in; (void)out_size; (void)ws_size;
  const float* pred   = (const float*)d_in[0];
  const int*   target = (const int*)d_in[1];
  float*       out    = (float*)d_out;

  const size_t SHW = (size_t)B_ * C_ * H_ * W_;   // 4,980,736 elems
  float* tmpA = (float*)d_ws;                     // ~19.9 MB
  float* tmpB = tmpA + SHW;                       // ~19.9 MB
  float* bsum = tmpB + SHW;                       // 1024 floats

  bl_pass1<<<B_ * H_, W_, 0, stream>>>(target, tmpA, tmpB);
  bl_pass2_loss<<<B_ * W_, H_, 0, stream>>>(pred, target, tmpA, tmpB, bsum);
  bl_reduce<<<1, 32, 0, stream>>>(bsum, out, B_ * W_,
                                  1.0f / (float)(B_ * H_ * W_));
}